// SoftPatternClassifier_31739808317657
// MI455X (gfx1250) — compile-verified
//
#include <hip/hip_runtime.h>
#include <hip/hip_bf16.h>

typedef __attribute__((ext_vector_type(2))) float v2f;
typedef __attribute__((ext_vector_type(8))) float v8f;

#define T_TOK 2048
#define WDIM  300
#define DDIM  2700
#define NPAT  50
#define NTILE_N ((DDIM + 15) / 16)   /* 169 */
#define NTILE_M (T_TOK / 16)         /* 128 */

// sigmoid via v_exp_f32 + v_add + v_rcp_f32 (avoids IEEE div_scale/div_fmas chain)
__device__ __forceinline__ float sigmoidf_(float x) {
  return __builtin_amdgcn_rcpf(1.0f + __expf(-x));
}

// -------------------------------------------------------------------------
// Kernel 1: S[t, d] = sigmoid( E[doc[t], :] . diags[d, :] + bias[d] )
// One wave per 16x16 output tile, fp32 WMMA 16x16x4, K = 300 = 75 steps.
// A operand: lane(0-15)=row m, VGPRs hold K = 2*half + {0,1}
// B operand: lane(0-15)=col n (diags row), same K split (B = diags^T)
// -------------------------------------------------------------------------
__global__ void gemm_sigmoid_wmma(const int* __restrict__ doc,
                                  const float* __restrict__ emb,
                                  const float* __restrict__ diags,
                                  const float* __restrict__ bias,
                                  float* __restrict__ S) {
  const int wavesPerBlock = blockDim.x >> 5;
  const int waveId = blockIdx.x * wavesPerBlock + (threadIdx.x >> 5);
  const int tileM = waveId / NTILE_N;
  const int tileN = waveId % NTILE_N;
  if (tileM >= NTILE_M) return;

  const int lane = threadIdx.x & 31;
  const int half = lane >> 4;   // 0: K+{0,1}, 1: K+{2,3}
  const int l16  = lane & 15;

  const int m = tileM * 16 + l16;
  const int arow = doc[m];
  const float* __restrict__ aptr = emb + (size_t)arow * WDIM + 2 * half;

  const int n  = tileN * 16 + l16;
  const int nc = (n < DDIM) ? n : (DDIM - 1);
  const float* __restrict__ bptr = diags + (size_t)nc * WDIM + 2 * half;

  v8f acc = {};
#pragma unroll 5
  for (int k = 0; k < WDIM; k += 4) {
    v2f a = *(const v2f*)(aptr + k);
    v2f b = *(const v2f*)(bptr + k);
    acc = __builtin_amdgcn_wmma_f32_16x16x4_f32(
        /*neg_a=*/false, a, /*neg_b=*/false, b,
        /*c_mod=*/(short)0, acc, /*reuse_a=*/false, /*reuse_b=*/false);
  }

  if (n < DDIM) {
    const float bv = bias[n];
#pragma unroll
    for (int r = 0; r < 8; ++r) {
      const int row = tileM * 16 + r + 8 * half;
      S[(size_t)row * DDIM + n] = sigmoidf_(acc[r] + bv);
    }
  }
}

// -------------------------------------------------------------------------
// Kernel 2: sequential pattern-automaton scan, one lane per pattern.
// 6 groups (p=2..7) run as 6 concurrent blocks of one kernel.
// -------------------------------------------------------------------------
template <int P>
__device__ void scan_group(const float* __restrict__ S, int start,
                           const float* __restrict__ eps_raw,   // [NPAT, P-1]
                           float eps_sc, float sls,
                           float* __restrict__ scores) {        // [NPAT]
  const int pat = threadIdx.x;
  if (pat >= NPAT) return;

  float eps[P - 1];
#pragma unroll
  for (int j = 0; j < P - 1; ++j)
    eps[j] = eps_sc * sigmoidf_(eps_raw[pat * (P - 1) + j]);

  float h[P];
  h[0] = 1.0f;
#pragma unroll
  for (int j = 1; j < P; ++j) h[j] = 0.0f;
  float sc = 0.0f;

  const float* __restrict__ base = S + start + pat * 2 * P;
  for (int t = 0; t < T_TOK; ++t) {
    const float* __restrict__ tm = base + (size_t)t * DDIM;
    // deep prefetch: addresses are h-independent, so pull 4 steps ahead
    if (t + 4 < T_TOK) __builtin_prefetch(tm + 4 * DDIM, 0, 1);

    // tm[0..P-1] = self-loop values, tm[P..2P-1] = word-transition values
    float hh[P];
    hh[0] = h[0];
#pragma unroll
    for (int j = 1; j < P; ++j) hh[j] = h[j] + h[j - 1] * eps[j - 1];

    float res[P];
    res[0] = 1.0f + sls * hh[0] * tm[0];
#pragma unroll
    for (int j = 1; j < P; ++j)
      res[j] = hh[j - 1] * tm[P + j - 1] + sls * hh[j] * tm[j];

#pragma unroll
    for (int j = 0; j < P; ++j) h[j] = res[j];
    sc += res[P - 1];
  }
  scores[pat] = sc;
}

__global__ void scan_all_kernel(const float* __restrict__ S,
                                const float* __restrict__ e2,
                                const float* __restrict__ e3,
                                const float* __restrict__ e4,
                                const float* __restrict__ e5,
                                const float* __restrict__ e6,
                                const float* __restrict__ e7,
                                const float* __restrict__ eps_scale_p,
                                const float* __restrict__ sls_p,
                                float* __restrict__ scores) {
  const float eps_sc = sigmoidf_(eps_scale_p[0]);
  const float sls    = sigmoidf_(sls_p[0]);
  switch (blockIdx.x) {
    case 0: scan_group<2>(S,    0, e2, eps_sc, sls, scores +   0); break;
    case 1: scan_group<3>(S,  200, e3, eps_sc, sls, scores +  50); break;
    case 2: scan_group<4>(S,  500, e4, eps_sc, sls, scores + 100); break;
    case 3: scan_group<5>(S,  900, e5, eps_sc, sls, scores + 150); break;
    case 4: scan_group<6>(S, 1400, e6, eps_sc, sls, scores + 200); break;
    case 5: scan_group<7>(S, 2000, e7, eps_sc, sls, scores + 250); break;
  }
}

// -------------------------------------------------------------------------
// Kernel 3: tiny MLP head: out = relu(s@w1 + b1) @ w2 + b2
// -------------------------------------------------------------------------
__global__ void mlp_kernel(const float* __restrict__ s,     // [300]
                           const float* __restrict__ w1,    // [300,100]
                           const float* __restrict__ b1,    // [100]
                           const float* __restrict__ w2,    // [100,2]
                           const float* __restrict__ b2,    // [2]
                           float* __restrict__ out) {       // [2]
  __shared__ float hsh[100];
  const int j = threadIdx.x;
  if (j < 100) {
    float acc = b1[j];
    for (int k = 0; k < 300; ++k) acc += s[k] * w1[k * 100 + j];
    hsh[j] = acc > 0.0f ? acc : 0.0f;
  }
  __syncthreads();
  if (j < 2) {
    float acc = b2[j];
    for (int k = 0; k < 100; ++k) acc += hsh[k] * w2[k * 2 + j];
    out[j] = acc;
  }
}

extern "C" void kernel_launch(void* const* d_in, const int* in_sizes, int n_in,
                              void* d_out, int out_size, void* d_ws, size_t ws_size,
                              hipStream_t stream) {
  const int*   doc       = (const int*)d_in[0];
  const float* emb       = (const float*)d_in[1];
  const float* diags     = (const float*)d_in[2];
  const float* bias      = (const float*)d_in[3];
  const float* e2        = (const float*)d_in[4];
  const float* e3        = (const float*)d_in[5];
  const float* e4        = (const float*)d_in[6];
  const float* e5        = (const float*)d_in[7];
  const float* e6        = (const float*)d_in[8];
  const float* e7        = (const float*)d_in[9];
  const float* eps_scale = (const float*)d_in[10];
  const float* sls       = (const float*)d_in[11];
  const float* w1        = (const float*)d_in[12];
  const float* b1        = (const float*)d_in[13];
  const float* w2        = (const float*)d_in[14];
  const float* b2        = (const float*)d_in[15];
  float* out = (float*)d_out;

  float* S      = (float*)d_ws;                       // [2048 * 2700] fp32
  float* scores = S + (size_t)T_TOK * DDIM;           // [300] fp32

  // 1) WMMA GEMM + sigmoid: 128*169 = 21632 tiles, 8 waves/block -> 2704 blocks
  const int totalTiles = NTILE_M * NTILE_N;
  const int wavesPerBlock = 8;
  const int blocks = (totalTiles + wavesPerBlock - 1) / wavesPerBlock;
  gemm_sigmoid_wmma<<<blocks, wavesPerBlock * 32, 0, stream>>>(doc, emb, diags, bias, S);

  // 2) sequential scans, 6 concurrent groups
  scan_all_kernel<<<6, 64, 0, stream>>>(S, e2, e3, e4, e5, e6, e7,
                                        eps_scale, sls, scores);

  // 3) MLP head
  mlp_kernel<<<1, 128, 0, stream>>>(scores, w1, b1, w2, b2, out);
}